// L2CFastEmbedding_40802189312712
// MI455X (gfx1250) — compile-verified
//
#include <hip/hip_runtime.h>
#include <math.h>

// L2C fast-embedding disambiguation + scoring, MI455X (gfx1250, wave32).
// One wave32 per example. f32 shuffle reductions for decision-critical dots,
// V_WMMA_F32_16X16X4_F32 (B = ones, A = elementwise products) for the final
// 10 sigmoid dot products.

typedef float v2f __attribute__((ext_vector_type(2)));
typedef float v4f __attribute__((ext_vector_type(4)));
typedef float v8f __attribute__((ext_vector_type(8)));

#define N_SENSES 3
#define DIM 128
#define CTX 10
#define NSTEP 4
#define EPS_MARGIN 0.1f
#define WAVES_PER_BLOCK 4
#define ROWSTRIDE 132                 // 128 + 4 pad floats: 132%64==4 -> conflict-free banks
#define LDS_PER_WAVE (16 * ROWSTRIDE) // 16 A-matrix rows of f32 products

__device__ __forceinline__ float wave_sum32(float p) {
#pragma unroll
    for (int off = 16; off >= 1; off >>= 1)
        p += __shfl_xor(p, off, 32);
    return p; // every lane holds the total
}

__device__ __forceinline__ int pick_sense(float d0, float d1, float d2) {
    // jax.lax.top_k tie-break: lower index wins ties -> strict '>' replacement
    float best = d0; int bi = 0;
    if (d1 > best) { best = d1; bi = 1; }
    if (d2 > best) { best = d2; bi = 2; }
    float second = (bi == 0) ? fmaxf(d1, d2) : (bi == 1) ? fmaxf(d0, d2) : fmaxf(d0, d1);
    return (best - second > EPS_MARGIN) ? bi : -1;
}

__global__ __launch_bounds__(32 * WAVES_PER_BLOCK)
void l2c_fast_embedding_kernel(const int* __restrict__ x,
                               const float* __restrict__ Wg,
                               const float* __restrict__ Ws,
                               float* __restrict__ out, int nb) {
    __shared__ float smem[WAVES_PER_BLOCK * LDS_PER_WAVE];
    const int lane = threadIdx.x & 31;
    const int wave = threadIdx.x >> 5;
    int ex = blockIdx.x * WAVES_PER_BLOCK + wave;
    const bool active = (ex < nb);
    if (!active) ex = nb - 1; // keep wave alive through the block barrier

    const int* xe = x + (size_t)ex * (CTX + 1);
    const int word = xe[0];
    int ctx[CTX];
#pragma unroll
    for (int j = 0; j < CTX; ++j) ctx[j] = xe[1 + j];

    // Per-lane chunk: lane l owns floats [4l .. 4l+3] of a 128-dim row.
    auto grow = [&](int w) -> v4f {
        return reinterpret_cast<const v4f*>(Wg + (size_t)w * DIM)[lane];
    };
    auto srow = [&](int w, int s) -> v4f {
        return reinterpret_cast<const v4f*>(Ws + ((size_t)w * N_SENSES + (size_t)s) * DIM)[lane];
    };
    auto dot4 = [&](v4f a, v4f b) -> float {
        return wave_sum32(a.x * b.x + a.y * b.y + a.z * b.z + a.w * b.w);
    };

    // Initial context vector = sum of global vectors of all 10 context words.
    v4f cv = {0.f, 0.f, 0.f, 0.f};
#pragma unroll
    for (int j = 0; j < CTX; ++j) cv += grow(ctx[j]);

    // Sequential disambiguation over the first 4 context positions.
    // Each step reads senses[i] before it is ever written -> prev is always -1.
    int senses[NSTEP];
#pragma unroll
    for (int i = 0; i < NSTEP; ++i) {
        const int cw = ctx[i];
        v4f s0 = srow(cw, 0), s1 = srow(cw, 1), s2 = srow(cw, 2);
        const float d0 = dot4(s0, cv), d1 = dot4(s1, cv), d2 = dot4(s2, cv);
        const int ns = pick_sense(d0, d1, d2); // wave-uniform
        senses[i] = ns;
        if (ns >= 0) {
            v4f ch = (ns == 0) ? s0 : (ns == 1) ? s1 : s2;
            cv += ch - grow(cw); // swap global vec for sense vec
        }
    }

    // Disambiguate the target word against the updated context vector.
    v4f wv;
    {
        v4f s0 = srow(word, 0), s1 = srow(word, 1), s2 = srow(word, 2);
        const float d0 = dot4(s0, cv), d1 = dot4(s1, cv), d2 = dot4(s2, cv);
        const int ws = pick_sense(d0, d1, d2);
        wv = (ws < 0) ? grow(word) : (ws == 0) ? s0 : (ws == 1) ? s1 : s2;
    }

    // Stage A-matrix rows into LDS: row j = cvec_j (*) wv elementwise (f32).
    float* my = smem + wave * LDS_PER_WAVE;
#pragma unroll
    for (int j = 0; j < CTX; ++j) {
        const int sj = (j < NSTEP) ? senses[j] : -1; // wave-uniform
        v4f cvec = (sj < 0) ? grow(ctx[j]) : srow(ctx[j], sj);
        v4f prod = cvec * wv;
        reinterpret_cast<v4f*>(my + (size_t)j * ROWSTRIDE)[lane] = prod;
    }
    {
        v4f z = {0.f, 0.f, 0.f, 0.f};
#pragma unroll
        for (int j = CTX; j < 16; ++j)
            reinterpret_cast<v4f*>(my + (size_t)j * ROWSTRIDE)[lane] = z;
    }
    __syncthreads(); // make all lanes' LDS rows visible

    // D = A x Ones (+C) over 32 K=4 chunks: D[m][*] = sum_k prod_m[k] = dot(cvec_m, wv).
    // A f32 16x4 layout (ISA 7.12.2): VGPR0 lanes0-15 K=0 / lanes16-31 K=2; VGPR1 K=1/K=3.
    const float* arow = my + (size_t)(lane & 15) * ROWSTRIDE + ((lane >> 4) << 1);
    v8f acc = {0.f, 0.f, 0.f, 0.f, 0.f, 0.f, 0.f, 0.f};
    const v2f bones = {1.0f, 1.0f}; // B all-ones: result independent of B layout
#pragma unroll
    for (int c = 0; c < 32; ++c) {
        v2f a = *reinterpret_cast<const v2f*>(arow + 4 * c);
        acc = __builtin_amdgcn_wmma_f32_16x16x4_f32(
            /*neg_a=*/false, a, /*neg_b=*/false, bones,
            /*c_mod=*/(short)0, acc, /*reuse_a=*/false, /*reuse_b=*/false);
    }

    // D layout: VGPR r = (M=r, N=lane) for lanes 0-15; (M=8+r, N=lane-16) for lanes 16-31.
    // All columns are identical. Sum sigmoid over rows 0..9 excluding row 5, /9.
    float part = 0.f;
    if (lane == 0) {
#pragma unroll
        for (int j = 0; j < 8; ++j) {
            if (j == 5) continue;
            part += 1.f / (1.f + __expf(-acc[j]));
        }
    } else if (lane == 16) {
        part = 1.f / (1.f + __expf(-acc[0])) + 1.f / (1.f + __expf(-acc[1]));
    }
    part += __shfl_xor(part, 16, 32);
    if (active && lane == 0) out[ex] = part * (1.0f / 9.0f);
}

extern "C" void kernel_launch(void* const* d_in, const int* in_sizes, int n_in,
                              void* d_out, int out_size, void* d_ws, size_t ws_size,
                              hipStream_t stream) {
    const int*   x  = (const int*)d_in[0];
    const float* Wg = (const float*)d_in[1];
    const float* Ws = (const float*)d_in[2];
    float*       o  = (float*)d_out;
    const int nb = in_sizes[0] / (CTX + 1);
    const int blocks = (nb + WAVES_PER_BLOCK - 1) / WAVES_PER_BLOCK;
    hipLaunchKernelGGL(l2c_fast_embedding_kernel, dim3(blocks),
                       dim3(32 * WAVES_PER_BLOCK), 0, stream, x, Wg, Ws, o, nb);
}